// GraphSAGEEncoder_43130061586686
// MI455X (gfx1250) — compile-verified
//
#include <hip/hip_runtime.h>
#include <hip/hip_bf16.h>

#define HID 128
#define LDS_STRIDE 130  // pad 128 -> 130: keeps (row*130 + even k)*4 8B-aligned, banks conflict-free

typedef __attribute__((ext_vector_type(2))) float v2f;
typedef __attribute__((ext_vector_type(8))) float v8f;

// ---------------------------------------------------------------------------
// h[i, f] = emb[x[i], f]
__global__ void sage_gather(const float* __restrict__ emb,
                            const int* __restrict__ x,
                            float* __restrict__ h, int total) {
    int idx = blockIdx.x * blockDim.x + threadIdx.x;
    if (idx >= total) return;
    int node = idx >> 7;
    int f    = idx & (HID - 1);
    h[idx] = emb[(size_t)x[node] * HID + f];
}

// zero a float buffer
__global__ void sage_zero(float* __restrict__ p, int total) {
    int idx = blockIdx.x * blockDim.x + threadIdx.x;
    if (idx < total) p[idx] = 0.0f;
}

// cnt[dst[e]] += 1
__global__ void sage_count(const int* __restrict__ dst,
                           float* __restrict__ cnt, int n_edges) {
    int e = blockIdx.x * blockDim.x + threadIdx.x;
    if (e < n_edges) atomicAdd(&cnt[dst[e]], 1.0f);
}

// agg[dst[e], f] += h[src[e], f]   (one thread per (edge, feature))
__global__ void sage_scatter(const float* __restrict__ h,
                             const int* __restrict__ src,
                             const int* __restrict__ dst,
                             float* __restrict__ agg, int n_edges) {
    int idx = blockIdx.x * blockDim.x + threadIdx.x;
    int e = idx >> 7;
    if (e >= n_edges) return;
    int f = idx & (HID - 1);
    float v = h[(size_t)src[e] * HID + f];
    atomicAdd(&agg[(size_t)dst[e] * HID + f], v);
}

// ---------------------------------------------------------------------------
// out[i,:] = relu( (agg[i,:]/max(cnt[i],1)) @ Wl^T + bl + h[i,:] @ Wr^T )
//
// Block = 256 threads = 8 waves. Block owns 16 rows; wave w owns cols
// [16w, 16w+16). fp32 WMMA 16x16x4, K-loop of 32 steps per GEMM, both GEMMs
// accumulate into one v8f C tile. Requires n_nodes % 16 == 0 (100000 = 6250*16).
__global__ void __launch_bounds__(256)
sage_gemm(const float* __restrict__ h,
          const float* __restrict__ agg,
          const float* __restrict__ cnt,
          const float* __restrict__ Wl,
          const float* __restrict__ bl,
          const float* __restrict__ Wr,
          float* __restrict__ out) {
    __shared__ float sMean[16 * LDS_STRIDE];
    __shared__ float sH[16 * LDS_STRIDE];

    const int row0 = blockIdx.x * 16;

    // Cooperative panel load: 16 x 128 mean (scaled) and h into LDS.
    for (int i = threadIdx.x; i < 16 * HID; i += 256) {
        int r = i >> 7;
        int k = i & (HID - 1);
        size_t g = (size_t)(row0 + r) * HID + k;
        float c   = cnt[row0 + r];
        float inv = 1.0f / fmaxf(c, 1.0f);
        sMean[r * LDS_STRIDE + k] = agg[g] * inv;
        sH[r * LDS_STRIDE + k]    = h[g];
    }
    __syncthreads();

    const int lane = threadIdx.x & 31;
    const int wave = threadIdx.x >> 5;
    const int col0 = wave * 16;

    const int m  = lane & 15;          // A row / B col / C col within tile
    const int kb = (lane >> 4) << 1;   // lane group selects K pair (0 or 2)

    const float* wlRow = Wl + (size_t)(col0 + m) * HID;  // B[k][n] = Wl[col0+n][k]
    const float* wrRow = Wr + (size_t)(col0 + m) * HID;

    v8f acc = {};

#pragma unroll 8
    for (int k0 = 0; k0 < HID; k0 += 4) {
        // mean @ Wl^T
        v2f a0 = *(const v2f*)&sMean[m * LDS_STRIDE + k0 + kb];
        v2f b0 = *(const v2f*)&wlRow[k0 + kb];
        acc = __builtin_amdgcn_wmma_f32_16x16x4_f32(
                  false, a0, false, b0, (short)0, acc, false, false);
        // h @ Wr^T
        v2f a1 = *(const v2f*)&sH[m * LDS_STRIDE + k0 + kb];
        v2f b1 = *(const v2f*)&wrRow[k0 + kb];
        acc = __builtin_amdgcn_wmma_f32_16x16x4_f32(
                  false, a1, false, b1, (short)0, acc, false, false);
    }

    // C layout: VGPR v, lane L -> element (M = v + 8*(L/16), N = L%16)
    const float bias  = bl[col0 + m];
    const int   rbase = (lane >> 4) << 3;
#pragma unroll
    for (int v = 0; v < 8; ++v) {
        float val = acc[v] + bias;
        val = fmaxf(val, 0.0f);
        out[(size_t)(row0 + rbase + v) * HID + col0 + m] = val;
    }
}

// ---------------------------------------------------------------------------
extern "C" void kernel_launch(void* const* d_in, const int* in_sizes, int n_in,
                              void* d_out, int out_size, void* d_ws, size_t ws_size,
                              hipStream_t stream) {
    const int*   x    = (const int*)d_in[0];
    const int*   ei   = (const int*)d_in[1];
    const float* emb  = (const float*)d_in[2];
    const float* Wl1  = (const float*)d_in[3];
    const float* bl1  = (const float*)d_in[4];
    const float* Wr1  = (const float*)d_in[5];
    const float* Wl2  = (const float*)d_in[6];
    const float* bl2  = (const float*)d_in[7];
    const float* Wr2  = (const float*)d_in[8];

    const int n_nodes = in_sizes[0];
    const int n_edges = in_sizes[1] / 2;
    const int* src = ei;
    const int* dst = ei + n_edges;

    float* h   = (float*)d_ws;                     // [N, 128]
    float* agg = h   + (size_t)n_nodes * HID;      // [N, 128]
    float* cnt = agg + (size_t)n_nodes * HID;      // [N]

    const int featTotal = n_nodes * HID;

    // h = emb[x]
    sage_gather<<<(featTotal + 255) / 256, 256, 0, stream>>>(emb, x, h, featTotal);

    const int zeroTotal = featTotal + n_nodes;
    const int rowTiles  = n_nodes / 16;  // 100000 / 16 = 6250

    // ---- layer 1: h <- relu(sage_conv(h)) (in-place: GEMM is row-local) ----
    sage_zero<<<(zeroTotal + 255) / 256, 256, 0, stream>>>(agg, zeroTotal);
    sage_count<<<(n_edges + 255) / 256, 256, 0, stream>>>(dst, cnt, n_edges);
    sage_scatter<<<(n_edges * HID + 255) / 256, 256, 0, stream>>>(h, src, dst, agg, n_edges);
    sage_gemm<<<rowTiles, 256, 0, stream>>>(h, agg, cnt, Wl1, bl1, Wr1, h);

    // ---- layer 2: d_out <- relu(sage_conv(h)) ----
    sage_zero<<<(zeroTotal + 255) / 256, 256, 0, stream>>>(agg, zeroTotal);
    sage_count<<<(n_edges + 255) / 256, 256, 0, stream>>>(dst, cnt, n_edges);
    sage_scatter<<<(n_edges * HID + 255) / 256, 256, 0, stream>>>(h, src, dst, agg, n_edges);
    sage_gemm<<<rowTiles, 256, 0, stream>>>(h, agg, cnt, Wl2, bl2, Wr2, (float*)d_out);
}